// DecoderLayer_70600672411682
// MI455X (gfx1250) — compile-verified
//
#include <hip/hip_runtime.h>

typedef __attribute__((ext_vector_type(2))) float v2f;
typedef __attribute__((ext_vector_type(8))) float v8f;

#define FEAT_IN  128
#define FEAT_OUT 64
#define IMPROVED_FILL 2.0f

// ---------------------------------------------------------------------------
// 1) zero the aggregation accumulator (= d_out first half) and init degrees
//    with the improved-self-loop fill value.
// ---------------------------------------------------------------------------
__global__ void gcn_init_kernel(float* __restrict__ acc, float* __restrict__ deg, int N) {
    int idx = blockIdx.x * blockDim.x + threadIdx.x;
    if (idx < N * FEAT_OUT) acc[idx] = 0.0f;
    if (idx < N) deg[idx] = IMPROVED_FILL;
}

// ---------------------------------------------------------------------------
// 2) degree accumulation at the target (col) nodes.
// ---------------------------------------------------------------------------
__global__ void gcn_degree_kernel(const long long* __restrict__ col,
                                  const float* __restrict__ ew,
                                  float* __restrict__ deg, int E) {
    int e = blockIdx.x * blockDim.x + threadIdx.x;
    if (e < E) {
        atomicAdd(&deg[(int)col[e]], ew[e]);
    }
}

// ---------------------------------------------------------------------------
// 3) deg -> d^{-1/2} in place.
// ---------------------------------------------------------------------------
__global__ void gcn_dinv_kernel(float* __restrict__ deg, int N) {
    int i = blockIdx.x * blockDim.x + threadIdx.x;
    if (i < N) {
        float d = deg[i];
        deg[i] = (d > 0.0f) ? rsqrtf(d) : 0.0f;
    }
}

// ---------------------------------------------------------------------------
// 4) xw = x @ W via fp32 WMMA (V_WMMA_F32_16X16X4_F32).
//    One wave per 16-row tile; 4 accumulators cover all 64 output columns.
//    Per ISA 7.12.2 (32-bit matrices, wave32):
//      A 16x4 : lanes 0-15 -> M=lane, K={k,k+1}; lanes 16-31 -> K={k+2,k+3}
//      B 4x16 : lanes 0-15 -> N=lane, K={k,k+1}; lanes 16-31 -> K={k+2,k+3}
//      C 16x16: VGPR j -> M = j (lanes 0-15) / j+8 (lanes 16-31), N = lane&15
// ---------------------------------------------------------------------------
__global__ __launch_bounds__(256) void gcn_gemm_wmma_kernel(
        const float* __restrict__ x, const float* __restrict__ W,
        float* __restrict__ xw, int M) {
    const int wave = threadIdx.x >> 5;
    const int lane = threadIdx.x & 31;
    const int tile = blockIdx.x * 8 + wave;          // 16-row tile index
    if (tile * 16 >= M) return;                      // wave-uniform exit (EXEC stays full)

    const int half = lane >> 4;                      // 0: K pair {k,k+1}; 1: {k+2,k+3}
    const int l16  = lane & 15;
    const int arow = tile * 16 + l16;

    v8f c0 = {}, c1 = {}, c2 = {}, c3 = {};

    #pragma unroll
    for (int k = 0; k < FEAT_IN; k += 4) {
        const int ka = k + half * 2;
        // A fragment: x[arow][ka], x[arow][ka+1]  (ka is even -> aligned float2)
        v2f a = *(const v2f*)(x + arow * FEAT_IN + ka);

        // B fragments for the 4 column tiles of W (stride-64 rows; W is cache-hot)
        const float* Wk0 = W + ka * FEAT_OUT;
        const float* Wk1 = W + (ka + 1) * FEAT_OUT;
        v2f b0, b1, b2, b3;
        b0.x = Wk0[l16];      b0.y = Wk1[l16];
        b1.x = Wk0[16 + l16]; b1.y = Wk1[16 + l16];
        b2.x = Wk0[32 + l16]; b2.y = Wk1[32 + l16];
        b3.x = Wk0[48 + l16]; b3.y = Wk1[48 + l16];

        c0 = __builtin_amdgcn_wmma_f32_16x16x4_f32(false, a, false, b0, (short)0, c0, false, false);
        c1 = __builtin_amdgcn_wmma_f32_16x16x4_f32(false, a, false, b1, (short)0, c1, false, false);
        c2 = __builtin_amdgcn_wmma_f32_16x16x4_f32(false, a, false, b2, (short)0, c2, false, false);
        c3 = __builtin_amdgcn_wmma_f32_16x16x4_f32(false, a, false, b3, (short)0, c3, false, false);
    }

    const int orow = tile * 16 + half * 8;
    #pragma unroll
    for (int j = 0; j < 8; ++j) {
        float* dst = xw + (orow + j) * FEAT_OUT + l16;
        dst[0]  = c0[j];
        dst[16] = c1[j];
        dst[32] = c2[j];
        dst[48] = c3[j];
    }
}

// ---------------------------------------------------------------------------
// 5) edge scatter: 64 threads per edge (edge is wave-uniform -> dinv/ew loads
//    broadcast), fp32 atomic adds into the accumulator.
// ---------------------------------------------------------------------------
__global__ void gcn_scatter_kernel(const long long* __restrict__ row,
                                   const long long* __restrict__ col,
                                   const float* __restrict__ ew,
                                   const float* __restrict__ dinv,
                                   const float* __restrict__ xw,
                                   float* __restrict__ acc, int E) {
    int gid = blockIdx.x * blockDim.x + threadIdx.x;
    int e = gid >> 6;          // edge id
    int f = gid & 63;          // feature id
    if (e < E) {
        int r = (int)row[e];
        int c = (int)col[e];
        float norm = dinv[r] * ew[e] * dinv[c];
        atomicAdd(&acc[c * FEAT_OUT + f], xw[r * FEAT_OUT + f] * norm);
    }
}

// ---------------------------------------------------------------------------
// 6) finalize: add self-loop term + bias, duplicate into both tuple halves.
// ---------------------------------------------------------------------------
__global__ void gcn_finalize_kernel(float* __restrict__ out0,
                                    float* __restrict__ out1,
                                    const float* __restrict__ xw,
                                    const float* __restrict__ dinv,
                                    const float* __restrict__ b, int N) {
    int idx = blockIdx.x * blockDim.x + threadIdx.x;
    if (idx < N * FEAT_OUT) {
        int i = idx >> 6;      // node
        int f = idx & 63;      // feature
        float di = dinv[i];
        float self_norm = IMPROVED_FILL * di * di;
        float v = out0[idx] + xw[idx] * self_norm + b[f];
        out0[idx] = v;
        out1[idx] = v;
    }
}

extern "C" void kernel_launch(void* const* d_in, const int* in_sizes, int n_in,
                              void* d_out, int out_size, void* d_ws, size_t ws_size,
                              hipStream_t stream) {
    const float*     x   = (const float*)d_in[0];          // [N,128]
    const long long* ei  = (const long long*)d_in[1];      // [2,E] int64
    const float*     ew  = (const float*)d_in[2];          // [E]
    const float*     W   = (const float*)d_in[3];          // [128,64]
    const float*     b   = (const float*)d_in[4];          // [64]

    const int N = in_sizes[0] / FEAT_IN;
    const int E = in_sizes[2];
    const long long* row = ei;
    const long long* col = ei + E;

    // workspace: xw [N*64] floats, then deg/dinv [N] floats
    float* xw   = (float*)d_ws;
    float* dinv = xw + (size_t)N * FEAT_OUT;

    float* out0 = (float*)d_out;                 // accumulator + first tuple half
    float* out1 = out0 + (size_t)N * FEAT_OUT;   // second tuple half

    const int B256 = 256;

    // 1) init acc + degrees
    {
        int total = N * FEAT_OUT;
        gcn_init_kernel<<<(total + B256 - 1) / B256, B256, 0, stream>>>(out0, dinv, N);
    }
    // 2) degree accumulation
    gcn_degree_kernel<<<(E + B256 - 1) / B256, B256, 0, stream>>>(col, ew, dinv, E);
    // 3) rsqrt
    gcn_dinv_kernel<<<(N + B256 - 1) / B256, B256, 0, stream>>>(dinv, N);
    // 4) GEMM via fp32 WMMA: one wave per 16-row tile, 8 waves per block
    {
        int mtiles = (N + 15) / 16;
        int blocks = (mtiles + 7) / 8;
        gcn_gemm_wmma_kernel<<<blocks, B256, 0, stream>>>(x, W, xw, N);
    }
    // 5) edge scatter (64 threads / edge)
    {
        long long total = (long long)E * FEAT_OUT;
        long long blocks = (total + B256 - 1) / B256;
        gcn_scatter_kernel<<<(int)blocks, B256, 0, stream>>>(row, col, ew, dinv, xw, out0, E);
    }
    // 6) self-loop + bias + duplicate
    {
        int total = N * FEAT_OUT;
        gcn_finalize_kernel<<<(total + B256 - 1) / B256, B256, 0, stream>>>(out0, out1, xw, dinv, b, N);
    }
}